// GCN_30691836297381
// MI455X (gfx1250) — compile-verified
//
#include <hip/hip_runtime.h>

typedef float v2f __attribute__((ext_vector_type(2)));
typedef float v8f __attribute__((ext_vector_type(8)));

#define N_NODES 16384
#define TILE_M 16
#define WAVES_PER_BLOCK 8
#define BLOCK_THREADS (WAVES_PER_BLOCK * 32)

// Computes a 16-row tile of  T = A[M,K] @ Y[K,16]  with f32 WMMA, 8 waves
// splitting K, LDS cross-wave reduction, then a per-MODE epilogue:
//   MODE 0: out[m][n] = T[m][n]                       (store [M,16])
//   MODE 1: out[m][n] = relu(T[m][n] + bias[n])       (store [M,16])
//   MODE 2: out[m][c] = sum_j T[m][j]*W2[j][c]+bias[c](store [M,4])
template <int MODE>
__global__ __launch_bounds__(BLOCK_THREADS) void gcn_gemm16(
    const float* __restrict__ A,     // [M x K] row-major
    const float* __restrict__ Y,     // [K x 16] row-major
    const float* __restrict__ bias,  // b1[16] (MODE 1) or b2[4] (MODE 2)
    const float* __restrict__ W2,    // [16 x 4] row-major (MODE 2)
    float* __restrict__ out,
    int K) {
  __shared__ float red[WAVES_PER_BLOCK * 256];

  const int lane = threadIdx.x & 31;
  const int wave = threadIdx.x >> 5;
  const int row_base = blockIdx.x * TILE_M;

  // ISA 7.12.2 "32-bit A-Matrix 16x4": lanes 0-15 -> M=0..15 (K=0,1),
  // lanes 16-31 -> M=0..15 (K=2,3).  B 4x16 mirrors it (row striped in lanes).
  const int m = lane & 15;             // A row within tile / B column
  const int koff = (lane >> 4) << 1;   // 0 or 2: first K element this lane holds

  const int kPerWave = K / WAVES_PER_BLOCK;
  const int k_begin = wave * kPerWave;

  const float* arow = A + (size_t)(row_base + m) * (size_t)K + k_begin + koff;
  const float* ycol = Y + (size_t)(k_begin + koff) * 16 + m;  // Y[k][n], n == m bits

  v8f c0 = {};
  v8f c1 = {};
#pragma unroll 2
  for (int k = 0; k < kPerWave; k += 8) {
    __builtin_prefetch(arow + k + 256, 0, 0);  // stream A ~1KB ahead per row
    v2f a0 = *(const v2f*)(arow + k);          // A[m][k..k+1] (+koff), b64 load
    v2f a1 = *(const v2f*)(arow + k + 4);
    v2f b0, b1f;
    b0.x = ycol[(k + 0) * 16];                 // Y[k+koff  ][n]
    b0.y = ycol[(k + 0) * 16 + 16];            // Y[k+koff+1][n]
    b1f.x = ycol[(k + 4) * 16];
    b1f.y = ycol[(k + 4) * 16 + 16];
    c0 = __builtin_amdgcn_wmma_f32_16x16x4_f32(false, a0, false, b0,
                                               (short)0, c0, false, false);
    c1 = __builtin_amdgcn_wmma_f32_16x16x4_f32(false, a1, false, b1f,
                                               (short)0, c1, false, false);
  }
#pragma unroll
  for (int r = 0; r < 8; ++r) c0[r] += c1[r];

  // C/D layout: VGPR r, lanes 0-15 -> M=r, N=lane; lanes 16-31 -> M=r+8.
  float* my = red + wave * 256;
  const int mhi = (lane >> 4) << 3;  // 0 or 8
#pragma unroll
  for (int r = 0; r < 8; ++r) my[(r + mhi) * 16 + (lane & 15)] = c0[r];
  __syncthreads();

  // Cross-wave reduction: thread t owns element (M=t/16, N=t%16).
  const int t = threadIdx.x;
  float sum = 0.0f;
#pragma unroll
  for (int w = 0; w < WAVES_PER_BLOCK; ++w) sum += red[w * 256 + t];

  const int tm = t >> 4;
  const int tn = t & 15;
  if (MODE == 0) {
    out[(size_t)(row_base + tm) * 16 + tn] = sum;
  } else if (MODE == 1) {
    out[(size_t)(row_base + tm) * 16 + tn] = fmaxf(sum + bias[tn], 0.0f);
  } else {
    red[t] = sum;  // each thread rewrites only its own slot (safe pre-barrier)
    __syncthreads();
    if (t < TILE_M * 4) {
      const int rM = t >> 2;
      const int cO = t & 3;
      float acc = bias[cO];
#pragma unroll
      for (int j = 0; j < 16; ++j) acc += red[rM * 16 + j] * W2[j * 4 + cO];
      out[(size_t)(row_base + rM) * 4 + cO] = acc;
    }
  }
}

extern "C" void kernel_launch(void* const* d_in, const int* in_sizes, int n_in,
                              void* d_out, int out_size, void* d_ws, size_t ws_size,
                              hipStream_t stream) {
  (void)in_sizes; (void)n_in; (void)out_size; (void)ws_size;
  const float* adjacency = (const float*)d_in[0];  // [16384,16384]
  const float* feature   = (const float*)d_in[1];  // [16384,128]
  const float* W1        = (const float*)d_in[2];  // [128,16]
  const float* b1        = (const float*)d_in[3];  // [16]
  const float* W2        = (const float*)d_in[4];  // [16,4]
  const float* b2        = (const float*)d_in[5];  // [4]
  float* out = (float*)d_out;                      // [16384,4]

  float* xw1 = (float*)d_ws;                       // [16384,16] = 1 MB
  float* h   = xw1 + (size_t)N_NODES * 16;         // [16384,16] = 1 MB

  dim3 grid(N_NODES / TILE_M);   // 1024 tiles
  dim3 block(BLOCK_THREADS);     // 8 wave32

  // 1) XW1 = X @ W1                      (K = 128)
  gcn_gemm16<0><<<grid, block, 0, stream>>>(feature, W1, nullptr, nullptr, xw1, 128);
  // 2) h = relu(A @ XW1 + b1)            (K = 16384, streams A: 1 GiB)
  gcn_gemm16<1><<<grid, block, 0, stream>>>(adjacency, xw1, b1, nullptr, h, N_NODES);
  // 3) out = (A @ h) @ W2 + b2           (K = 16384, streams A: 1 GiB)
  gcn_gemm16<2><<<grid, block, 0, stream>>>(adjacency, h, b2, W2, out, N_NODES);
}